// EncoderLayer_45423574122725
// MI455X (gfx1250) — compile-verified
//
#include <hip/hip_runtime.h>
#include <math.h>

#define D_MODEL 512
#define NHEAD 8
#define D_HEAD 64
#define C_OUT 64
#define SEQ_LEN 1024
#define PRED_LEN 256
#define TOPK 8
#define DIM_FF 2048
#define BATCH 8
#define NFREQ 512                 // rfft bins; f = 1..511 used
#define TOT_LEN (SEQ_LEN + PRED_LEN)
#define TWO_PI 6.28318530717958647692f

typedef float v2f __attribute__((ext_vector_type(2)));
typedef float v8f __attribute__((ext_vector_type(8)));

// ---------------------------------------------------------------------------
// 1) Direct DFT: Xf[b,d,f] for f in [1,512). One block per (b,d); the 1024-tap
//    sequence is staged in LDS. Phase uses exact integer reduction mod 1024.
// ---------------------------------------------------------------------------
__global__ __launch_bounds__(256)
void dft_kernel(const float* __restrict__ x, float* __restrict__ Xre,
                float* __restrict__ Xim) {
  int b = blockIdx.x / D_MODEL;
  int d = blockIdx.x % D_MODEL;
  __shared__ float xs[SEQ_LEN];
  for (int t = threadIdx.x; t < SEQ_LEN; t += 256)
    xs[t] = x[((size_t)b * SEQ_LEN + t) * D_MODEL + d];
  __syncthreads();
  for (int f = 1 + threadIdx.x; f < NFREQ; f += 256) {
    float re = 0.0f, im = 0.0f;
    for (int t = 0; t < SEQ_LEN; ++t) {
      int ft = (f * t) & (SEQ_LEN - 1);          // exact mod 1024
      float s, c;
      sincosf((TWO_PI / SEQ_LEN) * (float)ft, &s, &c);
      re = fmaf(xs[t], c, re);
      im = fmaf(xs[t], -s, im);                  // e^{-i2pi f t/T}
    }
    size_t o = ((size_t)b * D_MODEL + d) * NFREQ + f;
    Xre[o] = re;
    Xim[o] = im;
  }
}

// ---------------------------------------------------------------------------
// 2) Top-8 magnitude bins per (b,d); emit 2*|X|/T, phase, and int freq index.
// ---------------------------------------------------------------------------
__global__ __launch_bounds__(256)
void topk_kernel(const float* __restrict__ Xre, const float* __restrict__ Xim,
                 float* __restrict__ amp2, float* __restrict__ phs,
                 int* __restrict__ fidx) {
  int idx = blockIdx.x * 256 + threadIdx.x;
  if (idx >= BATCH * D_MODEL) return;
  const float* re = Xre + (size_t)idx * NFREQ;
  const float* im = Xim + (size_t)idx * NFREQ;
  int chosen[TOPK];
  for (int k = 0; k < TOPK; ++k) {
    float best = -1.0f;
    int bi = 1;
    for (int f = 1; f < NFREQ; ++f) {
      bool used = false;
      for (int j = 0; j < k; ++j) used |= (chosen[j] == f);
      if (used) continue;
      float r = re[f], i = im[f];
      float m = fmaf(r, r, i * i);
      if (m > best) { best = m; bi = f; }
    }
    chosen[k] = bi;
    float r = re[bi], i = im[bi];
    size_t o = (size_t)idx * TOPK + k;
    amp2[o] = 2.0f * sqrtf(fmaf(r, r, i * i)) / (float)SEQ_LEN;
    phs[o] = atan2f(i, r);
    fidx[o] = bi;
  }
}

// ---------------------------------------------------------------------------
// 3) Season reconstruction over 1280 steps, written straight to d_out slice.
// ---------------------------------------------------------------------------
__global__ __launch_bounds__(256)
void season_kernel(const float* __restrict__ amp2, const float* __restrict__ phs,
                   const int* __restrict__ fidx, float* __restrict__ season) {
  size_t gid = (size_t)blockIdx.x * 256 + threadIdx.x;
  if (gid >= (size_t)BATCH * TOT_LEN * D_MODEL) return;
  int d = (int)(gid % D_MODEL);
  int t = (int)((gid / D_MODEL) % TOT_LEN);
  int b = (int)(gid / ((size_t)D_MODEL * TOT_LEN));
  size_t base = ((size_t)b * D_MODEL + d) * TOPK;
  float s = 0.0f;
#pragma unroll
  for (int k = 0; k < TOPK; ++k) {
    int ft = (fidx[base + k] * t) & (SEQ_LEN - 1);     // exact mod 1024
    float ang = (TWO_PI / SEQ_LEN) * (float)ft + phs[base + k];
    s = fmaf(amp2[base + k], cosf(ang), s);
  }
  season[gid] = s;
}

// res2 = res - season[:, :SEQ_LEN]
__global__ __launch_bounds__(256)
void season_sub_kernel(const float* __restrict__ res,
                       const float* __restrict__ season,
                       float* __restrict__ res2) {
  size_t gid = (size_t)blockIdx.x * 256 + threadIdx.x;
  if (gid >= (size_t)BATCH * SEQ_LEN * D_MODEL) return;
  int d = (int)(gid % D_MODEL);
  int t = (int)((gid / D_MODEL) % SEQ_LEN);
  int b = (int)(gid / ((size_t)D_MODEL * SEQ_LEN));
  res2[gid] = res[gid] - season[((size_t)b * TOT_LEN + t) * D_MODEL + d];
}

// ---------------------------------------------------------------------------
// 4) fp32 WMMA GEMM:  C[M,N] = act( A[M,K] @ W[N,K]^T + bias )
//    256 threads = 8 waves; block tile 32x64 (2x4 waves of 16x16), BK=32.
//    float4 cooperative loads + register prefetch pipeline; fragments for a
//    whole K-tile are loaded before the 8 back-to-back V_WMMA_F32_16X16X4_F32.
//    LDS row stride = BK+4 floats (144B): 16B-aligned rows, conflict-free.
// ---------------------------------------------------------------------------
#define BM 32
#define BN 64
#define BK 32
#define LDP (BK + 4)

template <int ACT, bool GUARD>
__global__ __launch_bounds__(256)
void gemm_wmma_f32(const float* __restrict__ A, const float* __restrict__ W,
                   const float* __restrict__ bias, float* __restrict__ C,
                   int M, int N, int K) {
  __shared__ float As[BM][LDP];
  __shared__ float Ws[BN][LDP];
  int bm = blockIdx.x * BM;
  int bn = blockIdx.y * BN;
  int tid = threadIdx.x;
  int wave = tid >> 5, lane = tid & 31;
  int wm = (wave >> 2) * 16;            // 0 or 16
  int wn = (wave & 3) * 16;             // 0,16,32,48
  int r = lane & 15, hi = lane >> 4;

  // Cooperative-load coordinates: one float4 of A, two of W per thread.
  int lr = tid >> 3;                    // 0..31
  int lc = (tid & 7) << 2;              // 0,4,...,28

  const float4 z4 = make_float4(0.f, 0.f, 0.f, 0.f);
  auto ldA = [&](int k0) -> float4 {
    int gr = bm + lr;
    if (GUARD && gr >= M) return z4;
    return *(const float4*)&A[(size_t)gr * K + (k0 + lc)];
  };
  auto ldW = [&](int k0, int rowoff) -> float4 {
    return *(const float4*)&W[(size_t)(bn + lr + rowoff) * K + (k0 + lc)];
  };

  v8f acc = {};
  float4 pa = ldA(0);
  float4 pw0 = ldW(0, 0);
  float4 pw1 = ldW(0, 32);

  for (int k0 = 0; k0 < K; k0 += BK) {
    __syncthreads();                    // previous tile fully consumed
    *(float4*)&As[lr][lc] = pa;
    *(float4*)&Ws[lr][lc] = pw0;
    *(float4*)&Ws[lr + 32][lc] = pw1;
    __syncthreads();
    if (k0 + BK < K) {                  // prefetch next tile (overlaps WMMAs)
      pa = ldA(k0 + BK);
      pw0 = ldW(k0 + BK, 0);
      pw1 = ldW(k0 + BK, 32);
    }
    // Load all fragments for this K-tile, then issue WMMAs back-to-back.
    v2f af[8], bf[8];
#pragma unroll
    for (int kk = 0; kk < 8; ++kk) {
      af[kk] = *(const v2f*)&As[wm + r][4 * kk + 2 * hi];
      bf[kk] = *(const v2f*)&Ws[wn + r][4 * kk + 2 * hi];
    }
#pragma unroll
    for (int kk = 0; kk < 8; ++kk)
      acc = __builtin_amdgcn_wmma_f32_16x16x4_f32(false, af[kk], false, bf[kk],
                                                  (short)0, acc, false, false);
  }

  int col = bn + wn + r;
  float bv = (bias != nullptr) ? bias[col] : 0.0f;
#pragma unroll
  for (int i = 0; i < 8; ++i) {
    int row = bm + wm + 8 * hi + i;     // VGPR i: lanes0-15 -> M=i, 16-31 -> M=i+8
    if (!GUARD || row < M) {
      float v = acc[i] + bv;
      if (ACT == 1) v = 1.0f / (1.0f + expf(-v));
      C[(size_t)row * N + col] = v;
    }
  }
}

// ---------------------------------------------------------------------------
// 5) Growth: diff against z0 + exponential-smoothing scan, prepend v0.
//    sfull shape (B, SEQ_LEN+1, 512). One thread per (b, channel).
// ---------------------------------------------------------------------------
__global__ __launch_bounds__(256)
void growth_smooth_kernel(const float* __restrict__ vraw,
                          const float* __restrict__ gl_z0,
                          const float* __restrict__ gl_sw,
                          const float* __restrict__ gl_v0,
                          float* __restrict__ sfull) {
  int idx = blockIdx.x * 256 + threadIdx.x;
  if (idx >= BATCH * D_MODEL) return;
  int b = idx / D_MODEL, c = idx % D_MODEL, h = c / D_HEAD;
  float alpha = 1.0f / (1.0f + expf(-gl_sw[h]));
  float s = gl_v0[c];
  float prev = gl_z0[c];
  const float* vin = vraw + (size_t)b * SEQ_LEN * D_MODEL + c;
  float* out = sfull + (size_t)b * (SEQ_LEN + 1) * D_MODEL + c;
  out[0] = s;
  for (int t = 0; t < SEQ_LEN; ++t) {
    float cur = vin[(size_t)t * D_MODEL];
    float v = cur - prev;
    prev = cur;
    s = fmaf(alpha, s, (1.0f - alpha) * v);       // s = a*s + (1-a)*v
    out[(size_t)(t + 1) * D_MODEL] = s;
  }
}

// ---------------------------------------------------------------------------
// 6) Fused LayerNorm:  out[row] = LN( Xa[row] + sign * Xb[brow] ) * g + b
//    One block (256 thr) per row of 512; Xb row remapped for batch stride.
// ---------------------------------------------------------------------------
__global__ __launch_bounds__(256)
void ln_fused_kernel(const float* __restrict__ Xa, const float* __restrict__ Xb,
                     float sign, int b_rows_per_batch, int b_row_off,
                     const float* __restrict__ gam, const float* __restrict__ bet,
                     float* __restrict__ out) {
  int row = blockIdx.x;
  int b = row / SEQ_LEN, t = row % SEQ_LEN;
  int tid = threadIdx.x;
  const float* xa = Xa + (size_t)row * D_MODEL;
  const float* xb = Xb + ((size_t)b * b_rows_per_batch + t + b_row_off) * D_MODEL;
  float x0 = fmaf(sign, xb[tid], xa[tid]);
  float x1 = fmaf(sign, xb[tid + 256], xa[tid + 256]);
  __shared__ float ssum[256], ssq[256];
  ssum[tid] = x0 + x1;
  ssq[tid] = fmaf(x0, x0, x1 * x1);
  __syncthreads();
  for (int s = 128; s > 0; s >>= 1) {
    if (tid < s) { ssum[tid] += ssum[tid + s]; ssq[tid] += ssq[tid + s]; }
    __syncthreads();
  }
  float mean = ssum[0] * (1.0f / D_MODEL);
  float var = ssq[0] * (1.0f / D_MODEL) - mean * mean;
  float rstd = rsqrtf(var + 1e-5f);
  float* o = out + (size_t)row * D_MODEL;
  o[tid] = (x0 - mean) * rstd * gam[tid] + bet[tid];
  o[tid + 256] = (x1 - mean) * rstd * gam[tid + 256] + bet[tid + 256];
}

// ---------------------------------------------------------------------------
// 7) Level smoothing scan with aux growth:
//    s_t = a*s_{t-1} + (1-a)*(level_t - sproj_t) + a*gproj_t, s_{-1}=v0.
// ---------------------------------------------------------------------------
__global__ __launch_bounds__(256)
void level_smooth_kernel(const float* __restrict__ level,
                         const float* __restrict__ gp,
                         const float* __restrict__ sp,
                         const float* __restrict__ ll_sw,
                         const float* __restrict__ ll_v0,
                         float* __restrict__ out) {
  int idx = blockIdx.x * 256 + threadIdx.x;
  if (idx >= BATCH * C_OUT) return;
  int b = idx / C_OUT, c = idx % C_OUT;
  float alpha = 1.0f / (1.0f + expf(-ll_sw[c]));
  float s = ll_v0[c];
  size_t base = (size_t)b * SEQ_LEN * C_OUT + c;
  for (int t = 0; t < SEQ_LEN; ++t) {
    size_t o = base + (size_t)t * C_OUT;
    float l = level[o] - sp[o];
    s = alpha * s + (1.0f - alpha) * l + alpha * gp[o];
    out[o] = s;
  }
}

// ---------------------------------------------------------------------------
// Host-side orchestration
// ---------------------------------------------------------------------------
extern "C" void kernel_launch(void* const* d_in, const int* in_sizes, int n_in,
                              void* d_out, int out_size, void* d_ws,
                              size_t ws_size, hipStream_t stream) {
  (void)in_sizes; (void)n_in; (void)out_size; (void)ws_size;
  const float* res      = (const float*)d_in[0];
  const float* level    = (const float*)d_in[1];
  const float* gl_z0    = (const float*)d_in[2];
  const float* gl_in_w  = (const float*)d_in[3];
  const float* gl_in_b  = (const float*)d_in[4];
  const float* gl_sw    = (const float*)d_in[5];
  const float* gl_v0    = (const float*)d_in[6];
  const float* gl_out_w = (const float*)d_in[7];
  const float* gl_out_b = (const float*)d_in[8];
  const float* ll_sw    = (const float*)d_in[9];
  const float* ll_v0    = (const float*)d_in[10];
  const float* ll_gw    = (const float*)d_in[11];
  const float* ll_gb    = (const float*)d_in[12];
  const float* ll_sw2   = (const float*)d_in[13];
  const float* ll_sb    = (const float*)d_in[14];
  const float* ff_w1    = (const float*)d_in[15];
  const float* ff_w2    = (const float*)d_in[16];
  const float* n1_g     = (const float*)d_in[17];
  const float* n1_b     = (const float*)d_in[18];
  const float* n2_g     = (const float*)d_in[19];
  const float* n2_b     = (const float*)d_in[20];

  // Output tuple layout: res, level, growth, season (flat, in order).
  float* out_res    = (float*)d_out;
  float* out_level  = out_res + (size_t)BATCH * SEQ_LEN * D_MODEL;
  float* out_growth = out_level + (size_t)BATCH * SEQ_LEN * C_OUT;
  float* out_season = out_growth + (size_t)BATCH * (SEQ_LEN + 1) * D_MODEL;

  // Workspace carve-up (with reuse): ~130 MB total.
  char* ws = (char*)d_ws;
  size_t off = 0;
  auto alloc = [&](size_t nbytes) -> void* {
    void* p = (void*)(ws + off);
    off = (off + nbytes + 255) & ~(size_t)255;
    return p;
  };
  float* Xre   = (float*)alloc((size_t)BATCH * D_MODEL * NFREQ * 4);      // 8 MB
  float* Xim   = (float*)alloc((size_t)BATCH * D_MODEL * NFREQ * 4);      // 8 MB
  float* amp2  = (float*)alloc((size_t)BATCH * D_MODEL * TOPK * 4);
  float* phs   = (float*)alloc((size_t)BATCH * D_MODEL * TOPK * 4);
  int*   fidx  = (int*)  alloc((size_t)BATCH * D_MODEL * TOPK * 4);
  float* res2  = (float*)alloc((size_t)BATCH * SEQ_LEN * D_MODEL * 4);    // 16 MB
  float* vraw  = (float*)alloc((size_t)BATCH * SEQ_LEN * D_MODEL * 4);    // 16 MB
  float* sfull = (float*)alloc((size_t)BATCH * (SEQ_LEN + 1) * D_MODEL * 4); // 16.8 MB
  float* ffh   = (float*)alloc((size_t)BATCH * SEQ_LEN * DIM_FF * 4);     // 64 MB
  // Reuse dead regions:
  float* res3  = sfull;                 // sfull dead after growth GEMM
  float* ffout = vraw;                  // vraw dead after growth scan
  float* gproj = Xre;                   // spectrum dead after season_kernel
  float* sproj = Xre + (size_t)BATCH * SEQ_LEN * C_OUT;

  dim3 blk(256);

  // --- Season path ---
  dft_kernel<<<dim3(BATCH * D_MODEL), blk, 0, stream>>>(res, Xre, Xim);
  topk_kernel<<<dim3((BATCH * D_MODEL + 255) / 256), blk, 0, stream>>>(
      Xre, Xim, amp2, phs, fidx);
  season_kernel<<<dim3((BATCH * TOT_LEN * D_MODEL) / 256), blk, 0, stream>>>(
      amp2, phs, fidx, out_season);
  season_sub_kernel<<<dim3((BATCH * SEQ_LEN * D_MODEL) / 256), blk, 0, stream>>>(
      res, out_season, res2);

  const int Mrows = BATCH * SEQ_LEN;    // 8192

  // --- Growth path ---
  gemm_wmma_f32<0, false><<<dim3(Mrows / BM, D_MODEL / BN), blk, 0, stream>>>(
      res2, gl_in_w, gl_in_b, vraw, Mrows, D_MODEL, D_MODEL);
  growth_smooth_kernel<<<dim3((BATCH * D_MODEL + 255) / 256), blk, 0, stream>>>(
      vraw, gl_z0, gl_sw, gl_v0, sfull);
  {
    int Mg = BATCH * (SEQ_LEN + 1);     // 8200, needs row guards
    gemm_wmma_f32<0, true><<<dim3((Mg + BM - 1) / BM, D_MODEL / BN), blk, 0,
                             stream>>>(sfull, gl_out_w, gl_out_b, out_growth,
                                       Mg, D_MODEL, D_MODEL);
  }

  // --- res = LN1(res2 - growth[:,1:]) ---
  ln_fused_kernel<<<dim3(Mrows), blk, 0, stream>>>(
      res2, out_growth, -1.0f, SEQ_LEN + 1, 1, n1_g, n1_b, res3);

  // --- Feed-forward: sigmoid(res3 @ W1^T) @ W2^T ---
  gemm_wmma_f32<1, false><<<dim3(Mrows / BM, DIM_FF / BN), blk, 0, stream>>>(
      res3, ff_w1, nullptr, ffh, Mrows, DIM_FF, D_MODEL);
  gemm_wmma_f32<0, false><<<dim3(Mrows / BM, D_MODEL / BN), blk, 0, stream>>>(
      ffh, ff_w2, nullptr, ffout, Mrows, D_MODEL, DIM_FF);

  // --- res_out = LN2(res3 + ff) ---
  ln_fused_kernel<<<dim3(Mrows), blk, 0, stream>>>(
      res3, ffout, 1.0f, SEQ_LEN, 0, n2_g, n2_b, out_res);

  // --- Level path: per-batch 512->64 projections (growth/season slices are
  //     non-contiguous across batch, so launch per batch) ---
  for (int b = 0; b < BATCH; ++b) {
    gemm_wmma_f32<0, false><<<dim3(SEQ_LEN / BM, C_OUT / BN), blk, 0, stream>>>(
        out_growth + (size_t)b * (SEQ_LEN + 1) * D_MODEL, ll_gw, ll_gb,
        gproj + (size_t)b * SEQ_LEN * C_OUT, SEQ_LEN, C_OUT, D_MODEL);
    gemm_wmma_f32<0, false><<<dim3(SEQ_LEN / BM, C_OUT / BN), blk, 0, stream>>>(
        out_season + (size_t)b * TOT_LEN * D_MODEL, ll_sw2, ll_sb,
        sproj + (size_t)b * SEQ_LEN * C_OUT, SEQ_LEN, C_OUT, D_MODEL);
  }
  level_smooth_kernel<<<dim3((BATCH * C_OUT + 255) / 256), blk, 0, stream>>>(
      level, gproj, sproj, ll_sw, ll_v0, out_level);
}